// SpatialConv_19559281066401
// MI455X (gfx1250) — compile-verified
//
#include <hip/hip_runtime.h>
#include <stdint.h>

typedef float v2f __attribute__((ext_vector_type(2)));
typedef float v8f __attribute__((ext_vector_type(8)));

#define C_CH 128
#define H_DIM 36
#define W_DIM 100
#define HW    (H_DIM * W_DIM)          // 3600
#define CHW   (C_CH * HW)              // 460800
#define B_DIM 64
#define KW    9
#define PADK  4
#define PREP_PER_DIR (KW * 64 * 128 * 2)   // 147456 floats per direction

// ---- CDNA5 async global->LDS staging (ASYNCcnt path), via inline asm -------
__device__ __forceinline__ void async_load_b32(uint32_t ldsOff, const float* g) {
    asm volatile("global_load_async_to_lds_b32 %0, %1, off"
                 :: "v"(ldsOff), "v"(g) : "memory");
}
__device__ __forceinline__ void wait_asynccnt0() {
    asm volatile("s_wait_asynccnt 0x0" ::: "memory");
}

// ---------------------------------------------------------------------------
// Repack weights [c_out][c_in][k] -> [k][cin_pair][m][q] so a WMMA A-fragment
// (lane<16: kappa0,kappa0+1 ; lane>=16: kappa0+2,kappa0+3) is one coalesced
// 8-byte global load per lane.
// ---------------------------------------------------------------------------
__global__ __launch_bounds__(256) void prep_weights_kernel(
    const float* __restrict__ wd, const float* __restrict__ wu,
    const float* __restrict__ wr, const float* __restrict__ wl,
    float* __restrict__ wp)
{
    const int total = 4 * PREP_PER_DIR;
    for (int idx = blockIdx.x * blockDim.x + threadIdx.x; idx < total;
         idx += gridDim.x * blockDim.x) {
        int dir = idx / PREP_PER_DIR;
        int r   = idx - dir * PREP_PER_DIR;
        int k   = r / (64 * 128 * 2);
        int r2  = r - k * (64 * 128 * 2);
        int pair = r2 >> 8;
        int r3   = r2 & 255;
        int m    = r3 >> 1;
        int q    = r3 & 1;
        int cin  = pair * 2 + q;
        const float* w = (dir == 0) ? wd : (dir == 1) ? wu : (dir == 2) ? wr : wl;
        wp[idx] = w[(m * C_CH + cin) * KW + k];
    }
}

__global__ __launch_bounds__(256) void copy_kernel(
    const float4* __restrict__ src, float4* __restrict__ dst, int n4)
{
    int i = blockIdx.x * blockDim.x + threadIdx.x;
    if (i < n4) dst[i] = src[i];
}

// ---------------------------------------------------------------------------
// One scan step:  out[:, :, dstLine] += relu( conv1d(out[:, :, srcLine]) + b )
// GEMM per tap k: D[128,N] += W_k[128,128] * shift_k(src)[128,N] via
// V_WMMA_F32_16X16X4_F32 (K-dim = c_in, 32 K-slices per tap).
// Grid: (b, m-half, t-segment). 4 waves; wave = 16 c_out rows x NS col-tiles.
// LDS layout is kappa-PAIRED: lds_b[cin_pair][cpos][q] so each WMMA B
// fragment is one contiguous 8-byte LDS pair (-> ds_load_2addr_b64, no movs).
// Staged with per-element global_load_async_to_lds_b32 (ASYNCcnt), halo
// columns pre-zeroed with disjoint float4 DS stores.
// ---------------------------------------------------------------------------
template <int NS>
__global__ __launch_bounds__(128) void step_kernel(
    float* __restrict__ out,
    const float* __restrict__ wprep,   // already offset to this direction
    const float* __restrict__ bias,
    int srcLineOff, int dstLineOff, int tStride, int tLen)
{
    constexpr int PADL = NS * 16 + 8;
    __shared__ __align__(16) float lds_b[64 * 2 * PADL];

    const int tid  = threadIdx.x;
    const int b    = blockIdx.x;
    const int mh   = blockIdx.y;
    const int t0   = blockIdx.z * (NS * 16);
    const int wave = tid >> 5;
    const int lane = tid & 31;
    const int hi   = lane >> 4;
    const int ln   = lane & 15;
    const int m0   = mh * 64 + wave * 16;
    const size_t bOff = (size_t)b * CHW;
    const float* src  = out + bOff + srcLineOff;

    // Staged global-t range [g0, g1); LDS columns [cposA, cposB) are live.
    const int g0    = (t0 - PADK > 0) ? (t0 - PADK) : 0;
    int g1          = t0 + NS * 16 + PADK; if (g1 > tLen) g1 = tLen;
    const int cposA = g0 - t0 + PADK;
    const int cposB = g1 - t0 + PADK;

    // 1) zero halo columns; per pair-row the left/right halos are contiguous
    //    dword ranges [0, 2*cposA) and [2*cposB, 2*PADL)  (both /4 here).
    {
        const int lc4 = cposA >> 1;            // float4 count, left halo
        const int rc4 = (PADL - cposB) >> 1;   // float4 count, right halo
        const int per = lc4 + rc4;
        const float4 z = make_float4(0.f, 0.f, 0.f, 0.f);
        for (int j = tid; j < 64 * per; j += 128) {
            int row = j / per, o = j - row * per;
            int off = (o < lc4) ? (o << 2) : (2 * cposB + ((o - lc4) << 2));
            *reinterpret_cast<float4*>(&lds_b[row * (2 * PADL) + off]) = z;
        }
    }

    // 2) async-stage the live span of the source line into LDS (scatter b32)
    {
        const int cnt = g1 - g0;               // uniform per block
        for (int j = tid; j < C_CH * cnt; j += 128) {
            int cin = j / cnt, o = j - cin * cnt;
            int t   = g0 + o;
            uint32_t lo = (uint32_t)(uintptr_t)
                &lds_b[(cin >> 1) * (2 * PADL) + (t - t0 + PADK) * 2 + (cin & 1)];
            async_load_b32(lo, src + (size_t)cin * HW + (size_t)t * tStride);
        }
    }
    wait_asynccnt0();
    __syncthreads();

    // 3) WMMA main loop
    v8f acc[NS] = {};
    for (int k = 0; k < KW; ++k) {
        const float* wk = wprep + (size_t)k * (64 * 128 * 2);
        #pragma unroll 4
        for (int u = 0; u < 32; ++u) {
            // A fragment: one coalesced b64 per lane from repacked weights
            v2f afrag = *reinterpret_cast<const v2f*>(
                wk + ((size_t)(2 * u + hi) * 128 + m0 + ln) * 2);
            // B fragments: contiguous 8-byte pairs, s-tiles 128B apart
            const float* bp = lds_b + (2 * u + hi) * (2 * PADL) + (ln + k) * 2;
            #pragma unroll
            for (int s = 0; s < NS; ++s) {
                v2f bfrag = *reinterpret_cast<const v2f*>(bp + s * 32);
                acc[s] = __builtin_amdgcn_wmma_f32_16x16x4_f32(
                    false, afrag, false, bfrag, (short)0, acc[s], false, false);
            }
        }
    }

    // 4) fused epilogue: out[dst] += relu(acc + bias)
    float bv[8];
    #pragma unroll
    for (int v = 0; v < 8; ++v) bv[v] = bias[m0 + hi * 8 + v];

    #pragma unroll
    for (int s = 0; s < NS; ++s) {
        int t = t0 + s * 16 + ln;
        if (t < tLen) {
            #pragma unroll
            for (int v = 0; v < 8; ++v) {
                int m = m0 + hi * 8 + v;
                size_t idx = bOff + (size_t)m * HW + dstLineOff + (size_t)t * tStride;
                float r = acc[s][v] + bv[v];
                out[idx] += (r > 0.0f) ? r : 0.0f;
            }
        }
    }
}

// ---------------------------------------------------------------------------
extern "C" void kernel_launch(void* const* d_in, const int* in_sizes, int n_in,
                              void* d_out, int out_size, void* d_ws, size_t ws_size,
                              hipStream_t stream)
{
    const float* x   = (const float*)d_in[0];
    const float* w_d = (const float*)d_in[1];
    const float* b_d = (const float*)d_in[2];
    const float* w_u = (const float*)d_in[3];
    const float* b_u = (const float*)d_in[4];
    const float* w_r = (const float*)d_in[5];
    const float* b_r = (const float*)d_in[6];
    const float* w_l = (const float*)d_in[7];
    const float* b_l = (const float*)d_in[8];
    float* out = (float*)d_out;
    float* wp  = (float*)d_ws;   // 4 * 147456 floats = 2.36 MB of scratch

    prep_weights_kernel<<<576, 256, 0, stream>>>(w_d, w_u, w_r, w_l, wp);

    const int n4 = (B_DIM * CHW) / 4;
    copy_kernel<<<(n4 + 255) / 256, 256, 0, stream>>>(
        (const float4*)x, (float4*)out, n4);

    // (b, m-half, t-segment): 256 blocks x 4 waves = 1024 waves per step
    dim3 gridW(B_DIM, 2, 2);   // W-lines: segments [0,64) and [64,128) mask<100
    dim3 gridH(B_DIM, 2, 2);   // H-lines: segments [0,32) and [32,64)  mask<36

    // down pass: conv along W (tStride 1)
    for (int i = 1; i < H_DIM; ++i)
        step_kernel<4><<<gridW, 128, 0, stream>>>(
            out, wp + 0 * PREP_PER_DIR, b_d, (i - 1) * W_DIM, i * W_DIM, 1, W_DIM);

    // up pass
    for (int i = H_DIM - 2; i >= 1; --i)
        step_kernel<4><<<gridW, 128, 0, stream>>>(
            out, wp + 1 * PREP_PER_DIR, b_u, (i + 1) * W_DIM, i * W_DIM, 1, W_DIM);

    // right pass: conv along H (tStride W)
    for (int j = 1; j < W_DIM; ++j)
        step_kernel<2><<<gridH, 128, 0, stream>>>(
            out, wp + 2 * PREP_PER_DIR, b_r, j - 1, j, W_DIM, H_DIM);

    // left pass
    for (int j = W_DIM - 2; j >= 1; --j)
        step_kernel<2><<<gridH, 128, 0, stream>>>(
            out, wp + 3 * PREP_PER_DIR, b_l, j + 1, j, W_DIM, H_DIM);

    (void)in_sizes; (void)n_in; (void)out_size; (void)ws_size;
}